// MPSWithDense_7224134992122
// MI455X (gfx1250) — compile-verified
//
#include <hip/hip_runtime.h>

// MPS contraction for MI455X (gfx1250, wave32).
// Chain steps run as V_WMMA_F32_16X16X4_F32 with batch striped across lanes.
// K-chunk remap {2c,2c+1,2c+8,2c+9} lets the D-layout accumulator feed the
// next step's B operand directly (no cross-lane data movement in the loop).
// Weights are pre-swizzled once into the WMMA A-operand lane layout in d_ws
// so the hot loop does 4 coalesced global_load_b128 per step instead of 16
// strided global_load_b32 gathers.

typedef __attribute__((ext_vector_type(2))) float v2f;
typedef __attribute__((ext_vector_type(4))) float v4f;
typedef __attribute__((ext_vector_type(8))) float v8f;

#define WMMA4(A, Bm, C) \
  __builtin_amdgcn_wmma_f32_16x16x4_f32(false, (A), false, (Bm), (short)0, (C), false, false)

#define MPS_B    16384
#define MPS_F    512
#define MPS_HALF 256
#define NSIDE    255
#define TILES_PER_WAVE 2

// Prepacked layout: per side, v4f pre[s][c][lane]; lane quad = {a0.x,a0.y,a1.x,a1.y}
#define PRE_SIDE_V4F (NSIDE * 4 * 32)          // 32640 v4f per side
#define PRE_BYTES    ((size_t)2 * PRE_SIDE_V4F * 16)

// ---------------------------------------------------------------------------
// Pre-swizzle W_left / W_right into WMMA A-operand lane order.
// ---------------------------------------------------------------------------
__global__ __launch_bounds__(256) void mps_prepack_kernel(
    const float* __restrict__ Wl, const float* __restrict__ Wr,
    float* __restrict__ pre)
{
  const int tid = blockIdx.x * blockDim.x + threadIdx.x;
  if (tid >= 2 * NSIDE * 32) return;
  const int side = tid / (NSIDE * 32);
  const int rem  = tid % (NSIDE * 32);
  const int s    = rem / 32;
  const int lane = rem % 32;
  const int m    = lane & 15;
  const int hl   = lane >> 4;

  const float* Ws = (side ? Wr : Wl) + (long)s * 512;   // [d][p][e]: d*32+p*16+e
  v4f* dst = (v4f*)pre + (size_t)side * PRE_SIDE_V4F;

#pragma unroll
  for (int c = 0; c < 4; ++c) {
    const int r0 = 2 * c + 8 * hl;      // K-slot remap row pair {r0, r0+1}
    v4f q;
    if (side == 0) {                    // left chain:  A'[e=m, d] = W[d, p, e]
      q.x = Ws[(r0    ) * 32      + m];
      q.y = Ws[(r0 + 1) * 32      + m];
      q.z = Ws[(r0    ) * 32 + 16 + m];
      q.w = Ws[(r0 + 1) * 32 + 16 + m];
    } else {                            // right chain: A'[d=m, e] = W[d, p, e]
      q.x = Ws[m * 32      + r0    ];
      q.y = Ws[m * 32      + r0 + 1];
      q.z = Ws[m * 32 + 16 + r0    ];
      q.w = Ws[m * 32 + 16 + r0 + 1];
    }
    dst[((size_t)s * 4 + c) * 32 + lane] = q;
  }
}

// ---------------------------------------------------------------------------
// Shared pieces of the main kernel
// ---------------------------------------------------------------------------
__device__ __forceinline__ void mps_step(v8f& V, float xt, const v2f a0[4], const v2f a1[4]) {
  v8f P = {}; v8f Q = {};
#pragma unroll
  for (int c = 0; c < 4; ++c) {
    v2f bb; bb.x = V[2 * c]; bb.y = V[2 * c + 1];   // D-layout pair IS the B operand
    P = WMMA4(a0[c], bb, P);
    Q = WMMA4(a1[c], bb, Q);
  }
#pragma unroll
  for (int r = 0; r < 8; ++r) V[r] = fmaf(xt, Q[r], P[r]);
}

__device__ __forceinline__ void mps_tail(
    long b0, long b1, int lane, int m, int hl,
    const v8f& V0, const v8f& V1, const v8f& U0, const v8f& U1,
    const float* __restrict__ Wout, const float* __restrict__ d1w,
    const float* __restrict__ d1b,  const float* __restrict__ d2w,
    const float* __restrict__ d2b,  float* __restrict__ out)
{
  // fold dense head: y = c0 + sum_o weff[o] * h_o
  float weff[8] = {}, c0 = d2b[0];
#pragma unroll 4
  for (int h = 0; h < 24; ++h) {
    const float wh = d2w[h];
    c0 = fmaf(wh, d1b[h], c0);
#pragma unroll
    for (int o = 0; o < 8; ++o) weff[o] = fmaf(wh, d1w[h * 8 + o], weff[o]);
  }

  // output site: h_o[b] = sum_e (Wout[:,o,:]^T V)[e,b] * U[e,b]
  float y0 = c0, y1 = c0;
#pragma unroll
  for (int o = 0; o < 8; ++o) {
    v2f a[4];
#pragma unroll
    for (int c = 0; c < 4; ++c) {       // A'[e,d] = Wout[d,o,e], d = 2c+j+8*hl
      const int d0 = 2 * c + 8 * hl;
      a[c].x = Wout[(d0    ) * 128 + o * 16 + m];
      a[c].y = Wout[(d0 + 1) * 128 + o * 16 + m];
    }
    v8f G0 = {}, G1 = {};
#pragma unroll
    for (int c = 0; c < 4; ++c) {
      v2f bb0; bb0.x = V0[2 * c]; bb0.y = V0[2 * c + 1];
      v2f bb1; bb1.x = V1[2 * c]; bb1.y = V1[2 * c + 1];
      G0 = WMMA4(a[c], bb0, G0);
      G1 = WMMA4(a[c], bb1, G1);
    }
    float p0 = 0.f, p1 = 0.f;
#pragma unroll
    for (int r = 0; r < 8; ++r) { p0 = fmaf(G0[r], U0[r], p0); p1 = fmaf(G1[r], U1[r], p1); }
    p0 += __shfl_xor(p0, 16, 32);       // combine e=0..7 (low lanes) with e=8..15
    p1 += __shfl_xor(p1, 16, 32);
    y0 = fmaf(weff[o], p0, y0);
    y1 = fmaf(weff[o], p1, y1);
  }

  if (lane < 16) {
    out[b0] = y0;
    out[b1] = y1;
  }
}

// ---------------------------------------------------------------------------
// Fast path: weights pre-swizzled in d_ws, 4x global_load_b128 per step.
// ---------------------------------------------------------------------------
__global__ __launch_bounds__(256) void mps_dense_packed_kernel(
    const float* __restrict__ x,   const float* __restrict__ W0,
    const float* __restrict__ pre, const float* __restrict__ Wout,
    const float* __restrict__ WN,
    const float* __restrict__ d1w, const float* __restrict__ d1b,
    const float* __restrict__ d2w, const float* __restrict__ d2b,
    float* __restrict__ out)
{
  const int lane = threadIdx.x & 31;
  const int wave = blockIdx.x * (blockDim.x >> 5) + (threadIdx.x >> 5);
  const int m    = lane & 15;
  const int hl   = lane >> 4;

  const long b0 = (long)wave * (16 * TILES_PER_WAVE) + m;
  const long b1 = b0 + 16;

  const v4f* PL = (const v4f*)pre + lane;                    // left side
  const v4f* PR = (const v4f*)pre + PRE_SIDE_V4F + lane;     // right side

  // ---- init left vector Vt[d,b] ----
  v8f V0, V1;
  {
    const float xa = x[b0 * MPS_F];
    const float xb = x[b1 * MPS_F];
#pragma unroll
    for (int r = 0; r < 8; ++r) {
      const int d = r + 8 * hl;
      const float p0 = W0[d], p1 = W0[16 + d];
      V0[r] = fmaf(xa, p1, p0);
      V1[r] = fmaf(xb, p1, p0);
    }
  }

  // ---- left chain: sites t = 1 .. 255 ----
  for (int s = 0; s < NSIDE; ++s) {
    v2f a0[4], a1[4];
#pragma unroll
    for (int c = 0; c < 4; ++c) {
      const v4f q = PL[((size_t)s * 4 + c) * 32];            // one coalesced b128
      a0[c].x = q.x; a0[c].y = q.y;
      a1[c].x = q.z; a1[c].y = q.w;
    }
    const float xt0 = x[b0 * MPS_F + (s + 1)];
    const float xt1 = x[b1 * MPS_F + (s + 1)];
    mps_step(V0, xt0, a0, a1);
    mps_step(V1, xt1, a0, a1);
  }

  // ---- init right vector Ut[e,b] ----
  v8f U0, U1;
  {
    const float xa = x[b0 * MPS_F + (MPS_F - 1)];
    const float xb = x[b1 * MPS_F + (MPS_F - 1)];
#pragma unroll
    for (int r = 0; r < 8; ++r) {
      const int d = r + 8 * hl;
      const float p0 = WN[d * 2], p1 = WN[d * 2 + 1];
      U0[r] = fmaf(xa, p1, p0);
      U1[r] = fmaf(xb, p1, p0);
    }
  }

  // ---- right chain: s = 254 .. 0, x column 256+s ----
  for (int i = 0; i < NSIDE; ++i) {
    const int s = NSIDE - 1 - i;
    v2f a0[4], a1[4];
#pragma unroll
    for (int c = 0; c < 4; ++c) {
      const v4f q = PR[((size_t)s * 4 + c) * 32];
      a0[c].x = q.x; a0[c].y = q.y;
      a1[c].x = q.z; a1[c].y = q.w;
    }
    const int t = MPS_HALF + s;
    const float xt0 = x[b0 * MPS_F + t];
    const float xt1 = x[b1 * MPS_F + t];
    mps_step(U0, xt0, a0, a1);
    mps_step(U1, xt1, a0, a1);
  }

  mps_tail(b0, b1, lane, m, hl, V0, V1, U0, U1, Wout, d1w, d1b, d2w, d2b, out);
}

// ---------------------------------------------------------------------------
// Fallback (ws too small): gather weights directly from W_left / W_right.
// ---------------------------------------------------------------------------
__global__ __launch_bounds__(256) void mps_dense_kernel(
    const float* __restrict__ x,   const float* __restrict__ W0,
    const float* __restrict__ Wl,  const float* __restrict__ Wout,
    const float* __restrict__ Wr,  const float* __restrict__ WN,
    const float* __restrict__ d1w, const float* __restrict__ d1b,
    const float* __restrict__ d2w, const float* __restrict__ d2b,
    float* __restrict__ out)
{
  const int lane = threadIdx.x & 31;
  const int wave = blockIdx.x * (blockDim.x >> 5) + (threadIdx.x >> 5);
  const int m    = lane & 15;
  const int hl   = lane >> 4;

  const long b0 = (long)wave * (16 * TILES_PER_WAVE) + m;
  const long b1 = b0 + 16;

  v8f V0, V1;
  {
    const float xa = x[b0 * MPS_F];
    const float xb = x[b1 * MPS_F];
#pragma unroll
    for (int r = 0; r < 8; ++r) {
      const int d = r + 8 * hl;
      const float p0 = W0[d], p1 = W0[16 + d];
      V0[r] = fmaf(xa, p1, p0);
      V1[r] = fmaf(xb, p1, p0);
    }
  }

  for (int s = 0; s < NSIDE; ++s) {
    const float* Ws = Wl + (long)s * 512;
    v2f a0[4], a1[4];
#pragma unroll
    for (int c = 0; c < 4; ++c) {
      const int d0 = 2 * c + 8 * hl;
      a0[c].x = Ws[(d0    ) * 32      + m];
      a0[c].y = Ws[(d0 + 1) * 32      + m];
      a1[c].x = Ws[(d0    ) * 32 + 16 + m];
      a1[c].y = Ws[(d0 + 1) * 32 + 16 + m];
    }
    const float xt0 = x[b0 * MPS_F + (s + 1)];
    const float xt1 = x[b1 * MPS_F + (s + 1)];
    mps_step(V0, xt0, a0, a1);
    mps_step(V1, xt1, a0, a1);
  }

  v8f U0, U1;
  {
    const float xa = x[b0 * MPS_F + (MPS_F - 1)];
    const float xb = x[b1 * MPS_F + (MPS_F - 1)];
#pragma unroll
    for (int r = 0; r < 8; ++r) {
      const int d = r + 8 * hl;
      const float p0 = WN[d * 2], p1 = WN[d * 2 + 1];
      U0[r] = fmaf(xa, p1, p0);
      U1[r] = fmaf(xb, p1, p0);
    }
  }

  for (int i = 0; i < NSIDE; ++i) {
    const int s = NSIDE - 1 - i;
    const float* Ws = Wr + (long)s * 512;
    v2f a0[4], a1[4];
#pragma unroll
    for (int c = 0; c < 4; ++c) {
      const int e0 = 2 * c + 8 * hl;
      a0[c].x = Ws[m * 32      + e0    ];
      a0[c].y = Ws[m * 32      + e0 + 1];
      a1[c].x = Ws[m * 32 + 16 + e0    ];
      a1[c].y = Ws[m * 32 + 16 + e0 + 1];
    }
    const int t = MPS_HALF + s;
    const float xt0 = x[b0 * MPS_F + t];
    const float xt1 = x[b1 * MPS_F + t];
    mps_step(U0, xt0, a0, a1);
    mps_step(U1, xt1, a0, a1);
  }

  mps_tail(b0, b1, lane, m, hl, V0, V1, U0, U1, Wout, d1w, d1b, d2w, d2b, out);
}

extern "C" void kernel_launch(void* const* d_in, const int* in_sizes, int n_in,
                              void* d_out, int out_size, void* d_ws, size_t ws_size,
                              hipStream_t stream) {
  (void)in_sizes; (void)n_in; (void)out_size;
  const float* x    = (const float*)d_in[0];
  const float* W0   = (const float*)d_in[1];
  const float* Wl   = (const float*)d_in[2];
  const float* Wout = (const float*)d_in[3];
  const float* Wr   = (const float*)d_in[4];
  const float* WN   = (const float*)d_in[5];
  const float* d1w  = (const float*)d_in[6];
  const float* d1b  = (const float*)d_in[7];
  const float* d2w  = (const float*)d_in[8];
  const float* d2b  = (const float*)d_in[9];
  float* out = (float*)d_out;

  dim3 grid(64), block(256);   // 512 waves x 2 tiles = 1024 tiles of 16 batch rows

  if (d_ws != nullptr && ws_size >= PRE_BYTES) {
    // Re-swizzle weights every call (deterministic; no cross-call state).
    hipLaunchKernelGGL(mps_prepack_kernel, dim3(64), dim3(256), 0, stream,
                       Wl, Wr, (float*)d_ws);
    hipLaunchKernelGGL(mps_dense_packed_kernel, grid, block, 0, stream,
                       x, W0, (const float*)d_ws, Wout, WN,
                       d1w, d1b, d2w, d2b, out);
  } else {
    hipLaunchKernelGGL(mps_dense_kernel, grid, block, 0, stream,
                       x, W0, Wl, Wout, Wr, WN, d1w, d1b, d2w, d2b, out);
  }
}